// TRTHungarianLoss_88115549045382
// MI455X (gfx1250) — compile-verified
//
#include <hip/hip_runtime.h>
#include <hip/hip_bf16.h>
#include <math.h>

// Problem constants (must match reference)
#define Bn   64
#define Cn   32
#define Pn   5
#define TDn  8
#define NTn  8
#define NHn  32768
#define SCn  16
#define NCn  3

#define SEG_BLOCKS  1024
#define SEG_THREADS 256

typedef __attribute__((ext_vector_type(2))) float v2f;
typedef __attribute__((ext_vector_type(8))) float v8f;

// Full-wave (32 lane) sum using V_WMMA_F32_16X16X4_F32.
// A: VGPR0 = per-lane value (M=lane%16, K = lane<16?0:2), VGPR1 = 0.
// B = ones(4x16)  =>  D[m][n] = x[m] + x[m+16] for every n.
// Each lane's 8 accumulator regs hold 8 distinct rows (replicated over columns),
// so sum(c[0..7]) gives rows 0..7 (lanes<16) or rows 8..15 (lanes>=16);
// one xor-16 shuffle finishes the reduction. EXEC must be all ones.
__device__ __forceinline__ float wave32_sum(float x) {
  v2f a; a[0] = x;    a[1] = 0.0f;
  v2f b; b[0] = 1.0f; b[1] = 1.0f;
  v8f c = {};
  c = __builtin_amdgcn_wmma_f32_16x16x4_f32(false, a, false, b, (short)0, c,
                                            false, false);
  float s = c[0] + c[1] + c[2] + c[3] + c[4] + c[5] + c[6] + c[7];
  s += __shfl_xor(s, 16, 32);
  return s;
}

// ---------------------------------------------------------------------------
// Kernel 1: segmentation loss partials.  Streaming, HBM-bound.
// One row (16 f32) per lane per iteration; 4x b128 loads -> coalesced 2KB/wave.
// Per-block partial sum written to ws (deterministic two-pass reduction).
// ---------------------------------------------------------------------------
__global__ void seg_loss_kernel(const float* __restrict__ hit_logits,
                                const int* __restrict__ hit_labels,
                                float* __restrict__ partials) {
  const long total  = (long)Bn * (long)NHn;
  const long stride = (long)gridDim.x * blockDim.x;
  long r = (long)blockIdx.x * blockDim.x + threadIdx.x;

  float acc = 0.0f;
  for (; r < total; r += stride) {
    const float4* row = (const float4*)(hit_logits + r * SCn);
    if (r + stride < total)
      __builtin_prefetch(hit_logits + (r + stride) * SCn, 0, 0);
    float4 v0 = row[0], v1 = row[1], v2 = row[2], v3 = row[3];
    int lab = hit_labels[r];

    float m = fmaxf(fmaxf(fmaxf(v0.x, v0.y), fmaxf(v0.z, v0.w)),
                    fmaxf(fmaxf(v1.x, v1.y), fmaxf(v1.z, v1.w)));
    m = fmaxf(m, fmaxf(fmaxf(fmaxf(v2.x, v2.y), fmaxf(v2.z, v2.w)),
                       fmaxf(fmaxf(v3.x, v3.y), fmaxf(v3.z, v3.w))));
    float s = expf(v0.x - m) + expf(v0.y - m) + expf(v0.z - m) + expf(v0.w - m)
            + expf(v1.x - m) + expf(v1.y - m) + expf(v1.z - m) + expf(v1.w - m)
            + expf(v2.x - m) + expf(v2.y - m) + expf(v2.z - m) + expf(v2.w - m)
            + expf(v3.x - m) + expf(v3.y - m) + expf(v3.z - m) + expf(v3.w - m);

    float xl = v0.x;
    xl = (lab == 1)  ? v0.y : xl;  xl = (lab == 2)  ? v0.z : xl;
    xl = (lab == 3)  ? v0.w : xl;  xl = (lab == 4)  ? v1.x : xl;
    xl = (lab == 5)  ? v1.y : xl;  xl = (lab == 6)  ? v1.z : xl;
    xl = (lab == 7)  ? v1.w : xl;  xl = (lab == 8)  ? v2.x : xl;
    xl = (lab == 9)  ? v2.y : xl;  xl = (lab == 10) ? v2.z : xl;
    xl = (lab == 11) ? v2.w : xl;  xl = (lab == 12) ? v3.x : xl;
    xl = (lab == 13) ? v3.y : xl;  xl = (lab == 14) ? v3.z : xl;
    xl = (lab == 15) ? v3.w : xl;

    acc += m + logf(s) - xl;  // logsumexp(row) - row[label]
  }

  // wave fold (WMMA) + cross-wave fold in LDS
  float wsum = wave32_sum(acc);
  __shared__ float smem[SEG_THREADS / 32];
  int lane = threadIdx.x & 31;
  int wid  = threadIdx.x >> 5;
  if (lane == 0) smem[wid] = wsum;
  __syncthreads();
  if (threadIdx.x == 0) {
    float bsum = 0.0f;
#pragma unroll
    for (int i = 0; i < SEG_THREADS / 32; ++i) bsum += smem[i];
    partials[blockIdx.x] = bsum;
  }
}

// ---------------------------------------------------------------------------
// Kernel 2: per-sample Hungarian (Jonker-Volgenant, f64, mirrors reference
// exactly incl. transpose rule and first-min tie-break) + hungarian/label/
// vertex per-sample partials.  One thread per sample; trivially cheap.
// ---------------------------------------------------------------------------
__global__ void match_kernel(const float* __restrict__ pred_params,
                             const float* __restrict__ pred_logits,
                             const float* __restrict__ vertex,
                             const float* __restrict__ target_params,
                             const int* __restrict__ target_labels,
                             const int* __restrict__ preds_lengths,
                             const int* __restrict__ targets_lengths,
                             float* __restrict__ hung_out,
                             float* __restrict__ label_out,
                             float* __restrict__ vert_out) {
  int i = blockIdx.x * blockDim.x + threadIdx.x;
  if (i >= Bn) return;

  const float* pp = pred_params   + (long)i * Cn * Pn;
  const float* tp = target_params + (long)i * NTn * TDn;
  int pl = preds_lengths[i];
  int tl = targets_lengths[i];

  // cost[pred][target] in f64, L1 over 5 dims (target dims 3..7)
  double cost[Cn][NTn];
  for (int r = 0; r < pl; ++r)
    for (int c = 0; c < tl; ++c) {
      double s = 0.0;
      for (int d = 0; d < Pn; ++d)
        s += fabs((double)pp[r * Pn + d] - (double)tp[c * TDn + 3 + d]);
      cost[r][c] = s;
    }

  const bool tr = pl > tl;          // reference transposes if rows > cols
  const int n = tr ? tl : pl;       // n <= 8
  const int m = tr ? pl : tl;       // m <= 32
  const double INF = __builtin_inf();

  double u[NTn + 1], v[Cn + 1];
  int p[Cn + 1], way[Cn + 1];
  for (int j = 0; j <= n; ++j) u[j] = 0.0;
  for (int j = 0; j <= m; ++j) { v[j] = 0.0; p[j] = 0; way[j] = 0; }

  for (int ii = 1; ii <= n; ++ii) {
    p[0] = ii;
    int j0 = 0;
    double minv[Cn + 1];
    bool used[Cn + 1];
    for (int j = 0; j <= m; ++j) { minv[j] = INF; used[j] = false; }
    while (true) {
      used[j0] = true;
      int i0 = p[j0];
      double delta = INF;
      int j1 = -1;
      for (int j = 1; j <= m; ++j) {
        if (!used[j]) {
          double cr = tr ? cost[j - 1][i0 - 1] : cost[i0 - 1][j - 1];
          double cur = cr - u[i0] - v[j];
          if (cur < minv[j]) { minv[j] = cur; way[j] = j0; }
          if (minv[j] < delta) { delta = minv[j]; j1 = j; }  // first min wins
        }
      }
      for (int j = 0; j <= m; ++j) {
        if (used[j]) { u[p[j]] += delta; v[j] -= delta; }
        else         { minv[j] -= delta; }
      }
      j0 = j1;
      if (p[j0] == 0) break;
    }
    while (j0) { int j1 = way[j0]; p[j0] = p[j1]; j0 = j1; }
  }

  // Collect matches, hungarian L1, adjusted labels
  int adj[Cn];
  for (int c = 0; c < Cn; ++c) adj[c] = 1;
  double l1sum = 0.0;
  int k = 0;
  for (int j = 1; j <= m; ++j) {
    if (p[j]) {
      int rr = tr ? (j - 1) : (p[j] - 1);   // pred index
      int cc = tr ? (p[j] - 1) : (j - 1);   // target index
      float s = 0.0f;
      for (int d = 0; d < Pn; ++d)
        s += fabsf(pp[rr * Pn + d] - tp[cc * TDn + 3 + d]);
      l1sum += (double)s;
      adj[rr] = target_labels[i * NTn + cc];
      ++k;
    }
  }
  hung_out[i] = (float)(l1sum / (double)(k * Pn));

  // label NLL over NC=3, mean over C
  const float* lg = pred_logits + (long)i * Cn * NCn;
  float lsum = 0.0f;
  for (int c = 0; c < Cn; ++c) {
    float a = lg[c * NCn + 0], b = lg[c * NCn + 1], d = lg[c * NCn + 2];
    float mm = fmaxf(a, fmaxf(b, d));
    float lse = mm + logf(expf(a - mm) + expf(b - mm) + expf(d - mm));
    float sel = (adj[c] == 0) ? a : ((adj[c] == 1) ? b : d);
    lsum += lse - sel;
  }
  label_out[i] = lsum / (float)Cn;

  // vertex partial: sum over 3 dims of |v*w - t0*w|
  const float vw0 = 0.1f, vw1 = 0.1f, vw2 = 0.8f;
  float vs = fabsf(vertex[i * 3 + 0] * vw0 - tp[0] * vw0)
           + fabsf(vertex[i * 3 + 1] * vw1 - tp[1] * vw1)
           + fabsf(vertex[i * 3 + 2] * vw2 - tp[2] * vw2);
  vert_out[i] = vs;
}

// ---------------------------------------------------------------------------
// Kernel 3: finalize — deterministic reduction of all partials, weighted sum.
// One wave; WMMA fold for cross-lane sums.
// ---------------------------------------------------------------------------
__global__ void finalize_kernel(const float* __restrict__ ws,
                                float* __restrict__ out) {
  int lane = threadIdx.x;  // 32 threads
  float seg = 0.0f, hg = 0.0f, lb = 0.0f, vt = 0.0f;
  for (int j = lane; j < SEG_BLOCKS; j += 32) seg += ws[j];
  for (int j = lane; j < Bn; j += 32) {
    hg += ws[SEG_BLOCKS + j];
    lb += ws[SEG_BLOCKS + Bn + j];
    vt += ws[SEG_BLOCKS + 2 * Bn + j];
  }
  seg = wave32_sum(seg);
  hg  = wave32_sum(hg);
  lb  = wave32_sum(lb);
  vt  = wave32_sum(vt);
  if (lane == 0) {
    float seg_loss  = seg / ((float)NHn * (float)Bn);
    float hung_loss = hg / (float)Bn;
    float lab_loss  = lb / (float)Bn;
    float vert_loss = vt / (3.0f * (float)Bn);
    out[0] = hung_loss + lab_loss + vert_loss + seg_loss;  // weights all 1.0
  }
}

extern "C" void kernel_launch(void* const* d_in, const int* in_sizes, int n_in,
                              void* d_out, int out_size, void* d_ws,
                              size_t ws_size, hipStream_t stream) {
  const float* pred_params    = (const float*)d_in[0];
  const float* pred_logits    = (const float*)d_in[1];
  const float* vertex         = (const float*)d_in[2];
  const float* hit_logits     = (const float*)d_in[3];
  const float* target_params  = (const float*)d_in[4];
  const int*   target_labels  = (const int*)d_in[5];
  const int*   hit_labels     = (const int*)d_in[6];
  const int*   preds_lengths  = (const int*)d_in[7];
  const int*   targets_lengths= (const int*)d_in[8];

  float* ws = (float*)d_ws;
  float* seg_partials = ws;                     // [SEG_BLOCKS]
  float* hung_out     = ws + SEG_BLOCKS;        // [Bn]
  float* label_out    = ws + SEG_BLOCKS + Bn;   // [Bn]
  float* vert_out     = ws + SEG_BLOCKS + 2*Bn; // [Bn]

  seg_loss_kernel<<<SEG_BLOCKS, SEG_THREADS, 0, stream>>>(hit_logits,
                                                          hit_labels,
                                                          seg_partials);
  match_kernel<<<2, 32, 0, stream>>>(pred_params, pred_logits, vertex,
                                     target_params, target_labels,
                                     preds_lengths, targets_lengths,
                                     hung_out, label_out, vert_out);
  finalize_kernel<<<1, 32, 0, stream>>>(ws, (float*)d_out);
}